// YoloLossV7_16733192585449
// MI455X (gfx1250) — compile-verified
//
#include <hip/hip_runtime.h>
#include <math.h>

typedef float v2f __attribute__((ext_vector_type(2)));
typedef float v8f __attribute__((ext_vector_type(8)));

#define NBLK 128
#define MAXM 3072
#define NSC  3

__device__ __forceinline__ float sigf(float x) { return 1.0f / (1.0f + expf(-x)); }

__device__ __forceinline__ float bce_elem(float x, float t) {
    return fmaxf(x, 0.0f) - x * t + log1pf(expf(-fabsf(x)));
}

__device__ __forceinline__ float dev_ciou(float x1, float y1, float w1, float h1,
                                          float x2, float y2, float w2, float h2) {
    const float eps = 1e-7f;
    float b1x1 = x1 - w1 * 0.5f, b1x2 = x1 + w1 * 0.5f;
    float b1y1 = y1 - h1 * 0.5f, b1y2 = y1 + h1 * 0.5f;
    float b2x1 = x2 - w2 * 0.5f, b2x2 = x2 + w2 * 0.5f;
    float b2y1 = y2 - h2 * 0.5f, b2y2 = y2 + h2 * 0.5f;
    float iw = fmaxf(fminf(b1x2, b2x2) - fmaxf(b1x1, b2x1), 0.0f);
    float ih = fmaxf(fminf(b1y2, b2y2) - fmaxf(b1y1, b2y1), 0.0f);
    float inter = iw * ih;
    float uni = w1 * h1 + w2 * h2 - inter + eps;
    float iou = inter / uni;
    float cw = fmaxf(b1x2, b2x2) - fminf(b1x1, b2x1);
    float ch = fmaxf(b1y2, b2y2) - fminf(b1y1, b2y1);
    float c2 = cw * cw + ch * ch + eps;
    float dx = b2x1 + b2x2 - b1x1 - b1x2;
    float dy = b2y1 + b2y2 - b1y1 - b1y2;
    float rho2 = (dx * dx + dy * dy) * 0.25f;
    const float PI_F = 3.14159265358979323846f;
    float da = atanf(w2 / (h2 + eps)) - atanf(w1 / (h1 + eps));
    float v = (4.0f / (PI_F * PI_F)) * da * da;
    float alpha = v / (v - iou + (1.0f + eps));
    return iou - (rho2 / c2 + v * alpha);
}

// 256-lane block reduction using V_WMMA_F32_16X16X4_F32 (gfx1250).
// 4 accumulating WMMAs with an all-ones B (4x16) sum 64 LDS values each into
// the 16x16 f32 accumulator; final xor-16 shuffle folds rows 0-7 + 8-15.
// Valid on every lane of wave 0 (tid < 32 -> full wave, EXEC all-ones).
__device__ __forceinline__ float block_reduce_256(float v, float* sdata) {
    sdata[threadIdx.x] = v;
    __syncthreads();
    float out = 0.0f;
    if (threadIdx.x < 32) {
        v8f acc = {};
        v2f ones;
        ones.x = 1.0f;
        ones.y = 1.0f;
#pragma unroll
        for (int k = 0; k < 4; ++k) {
            v2f a;
            a.x = sdata[k * 64 + threadIdx.x * 2];
            a.y = sdata[k * 64 + threadIdx.x * 2 + 1];
            acc = __builtin_amdgcn_wmma_f32_16x16x4_f32(false, a, false, ones,
                                                        (short)0, acc, false, false);
        }
        float s = acc[0] + acc[1] + acc[2] + acc[3] + acc[4] + acc[5] + acc[6] + acc[7];
        s += __shfl_xor(s, 16, 32);
        out = s;
    }
    __syncthreads();
    return out;
}

__global__ void k_zero(float* p, int n) {
    int i = blockIdx.x * blockDim.x + threadIdx.x;
    int st = gridDim.x * blockDim.x;
    for (; i < n; i += st) p[i] = 0.0f;
}

// Serial target assignment (replicates reference ordering, incl. offset-major
// candidate order then anchor-major filtering). 192 targets x 5 offs x 3 a x 3
// scales -> negligible work; one lane keeps JAX .set duplicate semantics exact.
__global__ void k_build(const float* __restrict__ tgt, const float* __restrict__ imgsz,
                        float* cand, int* mi, float* mf, int* ncount) {
    if (threadIdx.x != 0 || blockIdx.x != 0) return;
    const int HS[3] = {80, 40, 20}, WSZ[3] = {80, 40, 20};
    const float ANCH[3][3][2] = {{{10, 13}, {16, 30}, {33, 23}},
                                 {{30, 61}, {62, 45}, {59, 119}},
                                 {{116, 90}, {156, 198}, {373, 326}}};
    const float OFF[5][2] = {{0, 0}, {1, 0}, {0, 1}, {-1, 0}, {0, -1}};
    const int F = NSC * MAXM;
    int *mb = mi, *ma = mi + F, *mgj = mi + 2 * F, *mgi = mi + 3 * F, *mcls = mi + 4 * F;
    float *tbx = mf, *tby = mf + F, *tbw = mf + 2 * F, *tbh = mf + 3 * F;
    float *faw = mf + 4 * F, *fah = mf + 5 * F;
    float is0 = imgsz[0], is1 = imgsz[1];
    for (int s = 0; s < 3; ++s) {
        int H = HS[s], W = WSZ[s];
        float gxg = (float)W / is1, gyg = (float)H / is0;
        float aw[3], ah[3];
        for (int a = 0; a < 3; ++a) {
            aw[a] = ANCH[s][a][0] * gxg;  // ANCHORS / stride_x
            ah[a] = ANCH[s][a][1] * gyg;  // ANCHORS / stride_y
        }
        int nc = 0;
        for (int o = 0; o < 5; ++o) {
            for (int r = 0; r < 192; ++r) {
                const float* T = tgt + r * 6;
                float cx = T[2] * gxg, cy = T[3] * gyg;
                float gw = T[4] * gxg, gh = T[5] * gyg;
                float gx = cx - OFF[o][0], gy = cy - OFF[o][1];
                if (gx >= 0.0f && gx < (float)W && gy >= 0.0f && gy < (float)H) {
                    float* C = cand + nc * 8;
                    C[0] = T[0]; C[1] = T[1]; C[2] = cx; C[3] = cy;
                    C[4] = gx;   C[5] = gy;   C[6] = gw; C[7] = gh;
                    ++nc;
                }
            }
        }
        int n = 0;
        for (int a = 0; a < 3; ++a) {
            for (int c = 0; c < nc; ++c) {
                const float* C = cand + c * 8;
                float rw = C[6] / aw[a], rh = C[7] / ah[a];
                float mx = fmaxf(fmaxf(rw, 1.0f / rw), fmaxf(rh, 1.0f / rh));
                if (mx < 4.0f && n < MAXM) {
                    int gi = (int)C[4], gj = (int)C[5];
                    int idx = s * MAXM + n;
                    mb[idx] = (int)C[0]; ma[idx] = a; mgj[idx] = gj; mgi[idx] = gi;
                    mcls[idx] = (int)C[1];
                    tbx[idx] = C[2] - (float)gi; tby[idx] = C[3] - (float)gj;
                    tbw[idx] = C[6]; tbh[idx] = C[7];
                    faw[idx] = aw[a]; fah[idx] = ah[a];
                    ++n;
                }
            }
        }
        ncount[s] = n;
    }
}

// Per-scale matched entries: CIoU (lbox), class BCE (lcls), ordered tobj writes.
__global__ void k_matched(const float* __restrict__ pi, int H, int W,
                          const int* __restrict__ mi, const float* __restrict__ mf,
                          int sOff, const int* __restrict__ ncount, int s,
                          float* tobj, float* iou_buf,
                          float* out_lbox, float* out_lcls) {
    __shared__ float sdata[256];
    const int F = NSC * MAXM;
    const int *mb = mi, *ma = mi + F, *mgj = mi + 2 * F, *mgi = mi + 3 * F, *mcls = mi + 4 * F;
    const float *tbx = mf, *tby = mf + F, *tbw = mf + 2 * F, *tbh = mf + 3 * F;
    const float *faw = mf + 4 * F, *fah = mf + 5 * F;
    int n = ncount[s];
    size_t HW = (size_t)H * W;
    float lb = 0.0f, lc = 0.0f;
    for (int m = threadIdx.x; m < n; m += blockDim.x) {
        int idx = sOff + m;
        int b = mb[idx], a = ma[idx], gj = mgj[idx], gi = mgi[idx], cls = mcls[idx];
        const float* base = pi + ((size_t)(b * 3 + a) * 85) * HW + (size_t)gj * W + gi;
        float px = sigf(base[0]) * 3.0f - 1.0f;
        float py = sigf(base[HW]) * 3.0f - 1.0f;
        float sw = sigf(base[2 * HW]) * 2.0f;
        float sh = sigf(base[3 * HW]) * 2.0f;
        float pw = sw * sw * faw[idx];
        float ph = sh * sh * fah[idx];
        float iou = dev_ciou(px, py, pw, ph, tbx[idx], tby[idx], tbw[idx], tbh[idx]);
        iou_buf[m] = iou;
        lb += 1.0f - iou;
        int tc = cls - 1;
        for (int c = 0; c < 80; ++c) {
            float x = base[(size_t)(5 + c) * HW];
            lc += bce_elem(x, (c == tc) ? 1.0f : 0.0f);
        }
    }
    float lbs = block_reduce_256(lb, sdata);
    float lcs = block_reduce_256(lc, sdata);
    if (threadIdx.x == 0) {
        *out_lbox = (n > 0) ? lbs / (float)n : 0.0f;
        *out_lcls = (n > 0) ? lcs / ((float)n * 80.0f) : 0.0f;
    }
    __syncthreads();
    if (threadIdx.x == 0) {  // in-order: last duplicate wins (matches .set)
        for (int m = 0; m < n; ++m) {
            int idx = sOff + m;
            size_t ti = (((size_t)mb[idx] * 3 + ma[idx]) * H + mgj[idx]) * W + mgi[idx];
            tobj[ti] = fmaxf(iou_buf[m], 0.0f);
        }
    }
}

// Objectness BCE over the full (B,NA,H,W) lattice; strided channel-4 gather
// with global_prefetch_b8, fixed-order WMMA block reduction -> per-block partial.
__global__ void k_lobj(const float* __restrict__ pi, const float* __restrict__ tobj,
                       int H, int W, float* partial) {
    __shared__ float sdata[256];
    size_t HW = (size_t)H * W;
    size_t N = (size_t)48 * HW;  // B(16) * NA(3) * H * W
    size_t stride = (size_t)gridDim.x * blockDim.x;
    float acc = 0.0f;
    for (size_t i = (size_t)blockIdx.x * blockDim.x + threadIdx.x; i < N; i += stride) {
        size_t ba = i / HW, hw = i - ba * HW;     // ba = b*3+a -> chan = ba*85+4
        size_t nx = i + stride;
        if (nx < N) {
            size_t ba2 = nx / HW, hw2 = nx - ba2 * HW;
            __builtin_prefetch(pi + (ba2 * 85 + 4) * HW + hw2, 0, 0);
        }
        float x = pi[(ba * 85 + 4) * HW + hw];
        acc += bce_elem(x, tobj[i]);
    }
    float s = block_reduce_256(acc, sdata);
    if (threadIdx.x == 0) partial[blockIdx.x] = s;
}

__global__ void k_final(const float* __restrict__ sums, const float* __restrict__ partials,
                        float* out) {
    if (threadIdx.x != 0 || blockIdx.x != 0) return;
    const float BAL[3] = {4.0f, 1.0f, 0.4f};
    const float CNT[3] = {16.0f * 3 * 80 * 80, 16.0f * 3 * 40 * 40, 16.0f * 3 * 20 * 20};
    float lobj = 0.0f;
    for (int s = 0; s < 3; ++s) {
        float t = 0.0f;
        for (int k = 0; k < NBLK; ++k) t += partials[s * NBLK + k];
        lobj += BAL[s] * t / CNT[s];
    }
    float lbox = (sums[0] + sums[1] + sums[2]) * 3.54f;
    float lcls = (sums[3] + sums[4] + sums[5]) * 37.4f;
    lobj *= 64.3f;
    out[0] = lbox + lobj + lcls;
    out[1] = lbox;
    out[2] = lobj;
    out[3] = lcls;
}

extern "C" void kernel_launch(void* const* d_in, const int* in_sizes, int n_in,
                              void* d_out, int out_size, void* d_ws, size_t ws_size,
                              hipStream_t stream) {
    const float* p[3] = {(const float*)d_in[0], (const float*)d_in[1], (const float*)d_in[2]};
    const float* targets = (const float*)d_in[3];
    const float* imgsz = (const float*)d_in[4];
    float* out = (float*)d_out;
    float* ws = (float*)d_ws;

    const int HS[3] = {80, 40, 20}, WSZ[3] = {80, 40, 20};
    const size_t tsz[3] = {16ull * 3 * 80 * 80, 16ull * 3 * 40 * 40, 16ull * 3 * 20 * 20};

    float* tobj0 = ws;
    float* tobj1 = tobj0 + tsz[0];
    float* tobj2 = tobj1 + tsz[1];
    float* tobj[3] = {tobj0, tobj1, tobj2};
    float* iou_buf = tobj2 + tsz[2];            // 3*MAXM floats
    float* cand = iou_buf + 3 * MAXM;           // 960*8 floats (reused per scale)
    int* mi = (int*)(cand + 960 * 8);           // 5*3*MAXM ints
    float* mf = (float*)(mi + 5 * 3 * MAXM);    // 6*3*MAXM floats
    int* ncount = (int*)(mf + 6 * 3 * MAXM);    // 4 ints
    float* sums = (float*)(ncount + 4);         // lbox[3], lcls[3]
    float* partials = sums + 6;                 // 3*NBLK floats

    int ztot = (int)(tsz[0] + tsz[1] + tsz[2]);
    k_zero<<<256, 256, 0, stream>>>(tobj0, ztot);
    k_build<<<1, 32, 0, stream>>>(targets, imgsz, cand, mi, mf, ncount);
    for (int s = 0; s < 3; ++s) {
        k_matched<<<1, 256, 0, stream>>>(p[s], HS[s], WSZ[s], mi, mf, s * MAXM,
                                         ncount, s, tobj[s], iou_buf + s * MAXM,
                                         sums + s, sums + 3 + s);
    }
    for (int s = 0; s < 3; ++s) {
        k_lobj<<<NBLK, 256, 0, stream>>>(p[s], tobj[s], HS[s], WSZ[s], partials + s * NBLK);
    }
    k_final<<<1, 32, 0, stream>>>(sums, partials, out);
}